// AttnDecoder_62955630625316
// MI455X (gfx1250) — compile-verified
//
#include <hip/hip_runtime.h>
#include <hip/hip_bf16.h>
#include <math.h>

// AttnDecoder greedy decode, MI455X (gfx1250, wave32).
// H=1024, A=1024, D=512, V=50257, S=128, K=A+H=2048, BOUND=25, SOS=1.
// Bandwidth-dominated by Wout (412 MB fp32 per step): done with
// V_WMMA_F32_16X16X4_F32 (exact fp32) streaming 16 rows per wave.

typedef __attribute__((ext_vector_type(2))) float v2f;
typedef __attribute__((ext_vector_type(8))) float v8f;

#define Hh   1024
#define Dd   512
#define Aa   1024
#define Vv   50257
#define Ss   128
#define KK   2048
#define BOUND 25
#define SOS  1

__device__ __forceinline__ float wave_sum(float v) {
#pragma unroll
  for (int o = 16; o > 0; o >>= 1) v += __shfl_xor(v, o, 32);
  return v;
}

// ---- init: h0 = hidden[-1], word = SOS -------------------------------------
__global__ void k_init(const float* __restrict__ hidden, float* __restrict__ h0,
                       int* __restrict__ word) {
  int t = threadIdx.x;
  if (t < Hh) h0[t] = hidden[t];
  if (t == 0) *word = SOS;
}

// ---- GRU cell: one wave per hidden unit j ----------------------------------
__global__ void k_gru(const float* __restrict__ E, const float* __restrict__ Wih,
                      const float* __restrict__ Whh, const float* __restrict__ bih,
                      const float* __restrict__ bhh, const int* __restrict__ word,
                      const float* __restrict__ hprev, float* __restrict__ hnew,
                      float* __restrict__ merge) {
  int lane = threadIdx.x & 31;
  int j = (int)((blockIdx.x * blockDim.x + threadIdx.x) >> 5);
  if (j >= Hh) return;
  const float* x = E + (size_t)(*word) * Dd;
  const float* wr = Wih + (size_t)j * Dd;
  const float* wz = Wih + (size_t)(j + Hh) * Dd;
  const float* wn = Wih + (size_t)(j + 2 * Hh) * Dd;
  float gir = 0.f, giz = 0.f, gin = 0.f;
  for (int k = lane; k < Dd; k += 32) {
    float xv = x[k];
    gir += wr[k] * xv; giz += wz[k] * xv; gin += wn[k] * xv;
  }
  const float* ur = Whh + (size_t)j * Hh;
  const float* uz = Whh + (size_t)(j + Hh) * Hh;
  const float* un = Whh + (size_t)(j + 2 * Hh) * Hh;
  float ghr = 0.f, ghz = 0.f, ghn = 0.f;
  for (int k = lane; k < Hh; k += 32) {
    float hv = hprev[k];
    ghr += ur[k] * hv; ghz += uz[k] * hv; ghn += un[k] * hv;
  }
  gir = wave_sum(gir); giz = wave_sum(giz); gin = wave_sum(gin);
  ghr = wave_sum(ghr); ghz = wave_sum(ghz); ghn = wave_sum(ghn);
  if (lane == 0) {
    float r = 1.f / (1.f + expf(-(gir + bih[j] + ghr + bhh[j])));
    float z = 1.f / (1.f + expf(-(giz + bih[j + Hh] + ghz + bhh[j + Hh])));
    float n = tanhf(gin + bih[j + 2 * Hh] + r * (ghn + bhh[j + 2 * Hh]));
    float hv = (1.f - z) * n + z * hprev[j];
    hnew[j] = hv;
    merge[j] = hv;          // merge[0:H] = h_new
  }
}

// ---- q = Wq . h_new : one wave per output a --------------------------------
__global__ void k_q(const float* __restrict__ Wq, const float* __restrict__ h,
                    float* __restrict__ q) {
  int lane = threadIdx.x & 31;
  int a = (int)((blockIdx.x * blockDim.x + threadIdx.x) >> 5);
  if (a >= Aa) return;
  const float* wr = Wq + (size_t)a * Hh;
  float acc = 0.f;
  for (int k = lane; k < Hh; k += 32) acc += wr[k] * h[k];
  acc = wave_sum(acc);
  if (lane == 0) q[a] = acc;
}

// ---- attention: scores, softmax, attn; writes weights[t] and merge[H:] -----
__global__ void k_attn(const float* __restrict__ emb, const float* __restrict__ q,
                       float* __restrict__ merge, float* __restrict__ w_out) {
  __shared__ float qs[Aa];
  __shared__ float sc[Ss];
  int tid = threadIdx.x, lane = tid & 31, wid = tid >> 5;   // 8 waves
  for (int i = tid; i < Aa; i += 256) qs[i] = q[i];
  __syncthreads();
  for (int s = wid; s < Ss; s += 8) {
    const float* er = emb + (size_t)s * Aa;
    float acc = 0.f;
    for (int k = lane; k < Aa; k += 32) acc += er[k] * qs[k];
    acc = wave_sum(acc);
    if (lane == 0) sc[s] = acc;
  }
  __syncthreads();
  if (tid == 0) {
    float mx = -3.4e38f;
    for (int s = 0; s < Ss; ++s) mx = fmaxf(mx, sc[s]);
    float sum = 0.f;
    for (int s = 0; s < Ss; ++s) { float e = expf(sc[s] - mx); sc[s] = e; sum += e; }
    float inv = 1.f / sum;
    for (int s = 0; s < Ss; ++s) sc[s] *= inv;
  }
  __syncthreads();
  for (int s = tid; s < Ss; s += 256) w_out[s] = sc[s];
  for (int a = tid; a < Aa; a += 256) {
    float acc = 0.f;
#pragma unroll 4
    for (int s = 0; s < Ss; ++s) acc += sc[s] * emb[(size_t)s * Aa + a];
    merge[Hh + a] = acc;    // merge[H:H+A] = attn
  }
}

// ---- Wout GEMV via V_WMMA_F32_16X16X4_F32: 16 rows per wave ----------------
// A tile = 16 Wout rows x 4 K (exact fp32); B tile = merge chunk broadcast to
// all 16 columns -> every D column equals the GEMV result for those 16 rows.
__global__ void k_wout_wmma(const float* __restrict__ Wout, const float* __restrict__ bout,
                            const float* __restrict__ merge, float* __restrict__ logits) {
  __shared__ float sm[KK];                       // 8 KB merge vector
  for (int i = threadIdx.x; i < KK; i += blockDim.x) sm[i] = merge[i];
  __syncthreads();
  int wid = threadIdx.x >> 5;
  int lane = threadIdx.x & 31;
  int group = blockIdx.x * (blockDim.x >> 5) + wid;
  int row_base = group * 16;
  if (row_base >= Vv) return;                    // whole-wave uniform exit
  int rl = lane & 15;
  int ksub = (lane >> 4) << 1;                   // lanes 0-15 -> K+{0,1}; 16-31 -> K+{2,3}
  int row = row_base + rl;
  if (row > Vv - 1) row = Vv - 1;                // clamp loads for the ragged tail
  const float* wrow = Wout + (size_t)row * KK + ksub;
  v8f c = {};
#pragma unroll 4
  for (int k0 = 0; k0 < KK; k0 += 4) {
    v2f a = *(const v2f*)(wrow + k0);            // global_load_b64, fully consumed lines
    v2f b; b.x = sm[k0 + ksub]; b.y = sm[k0 + ksub + 1];
    c = __builtin_amdgcn_wmma_f32_16x16x4_f32(
        /*neg_a=*/false, a, /*neg_b=*/false, b,
        /*c_mod=*/(short)0, c, /*reuse_a=*/false, /*reuse_b=*/false);
  }
  // D layout: lane 0 holds rows 0-7 (VGPR 0-7), lane 16 holds rows 8-15.
  if (lane == 0) {
#pragma unroll
    for (int i = 0; i < 8; ++i) {
      int r = row_base + i;
      if (r < Vv) logits[r] = c[i] + bout[r];
    }
  } else if (lane == 16) {
#pragma unroll
    for (int i = 0; i < 8; ++i) {
      int r = row_base + 8 + i;
      if (r < Vv) logits[r] = c[i] + bout[r];
    }
  }
}

// ---- log_softmax (in place) + argmax(first-max) + token feedback -----------
__global__ void k_finalize(float* __restrict__ logits, float* __restrict__ tok_out,
                           int* __restrict__ word_next) {
  __shared__ float sred[32];
  __shared__ int   sarg[32];
  __shared__ float s_M, s_L;
  __shared__ int   s_A;
  int tid = threadIdx.x, lane = tid & 31, wid = tid >> 5;   // 32 waves
  float m = -3.4e38f; int arg = 0;
  for (int v = tid; v < Vv; v += 1024) {
    float x = logits[v];
    if (x > m) { m = x; arg = v; }                // strided scan keeps first index
  }
#pragma unroll
  for (int o = 16; o > 0; o >>= 1) {
    float om = __shfl_xor(m, o, 32);
    int oa = __shfl_xor(arg, o, 32);
    if (om > m || (om == m && oa < arg)) { m = om; arg = oa; }
  }
  if (lane == 0) { sred[wid] = m; sarg[wid] = arg; }
  __syncthreads();
  if (wid == 0) {
    m = sred[lane]; arg = sarg[lane];
#pragma unroll
    for (int o = 16; o > 0; o >>= 1) {
      float om = __shfl_xor(m, o, 32);
      int oa = __shfl_xor(arg, o, 32);
      if (om > m || (om == m && oa < arg)) { m = om; arg = oa; }
    }
    if (lane == 0) { s_M = m; s_A = arg; }
  }
  __syncthreads();
  float M = s_M;
  float s = 0.f;
  for (int v = tid; v < Vv; v += 1024) s += expf(logits[v] - M);
  s = wave_sum(s);
  if (lane == 0) sred[wid] = s;
  __syncthreads();
  if (wid == 0) {
    s = wave_sum(sred[lane]);
    if (lane == 0) s_L = M + logf(s);
  }
  __syncthreads();
  float lse = s_L;
  for (int v = tid; v < Vv; v += 1024) logits[v] = logits[v] - lse;
  if (tid == 0) { *tok_out = (float)s_A; *word_next = s_A; }
}

extern "C" void kernel_launch(void* const* d_in, const int* in_sizes, int n_in,
                              void* d_out, int out_size, void* d_ws, size_t ws_size,
                              hipStream_t stream) {
  const float* hidden = (const float*)d_in[0];
  const float* emb    = (const float*)d_in[1];
  const float* E      = (const float*)d_in[2];
  const float* Wih    = (const float*)d_in[3];
  const float* Whh    = (const float*)d_in[4];
  const float* bih    = (const float*)d_in[5];
  const float* bhh    = (const float*)d_in[6];
  const float* Wq     = (const float*)d_in[7];
  const float* Wout   = (const float*)d_in[8];
  const float* bout   = (const float*)d_in[9];

  float* out  = (float*)d_out;
  float* lp   = out;                              // [25, V]
  float* wts  = out + (size_t)BOUND * Vv;         // [25, S]
  float* toks = wts + (size_t)BOUND * Ss;         // [25]

  float* ws    = (float*)d_ws;
  float* hbuf0 = ws;                              // 1024
  float* hbuf1 = ws + 1024;                       // 1024
  float* merge = ws + 2048;                       // 2048
  float* qbuf  = ws + 4096;                       // 1024
  int*   word  = (int*)(ws + 5120);               // 1

  k_init<<<1, 1024, 0, stream>>>(hidden, hbuf0, word);

  for (int t = 0; t < BOUND; ++t) {
    const float* hprev = (t & 1) ? hbuf1 : hbuf0;
    float*       hnew  = (t & 1) ? hbuf0 : hbuf1;

    k_gru<<<128, 256, 0, stream>>>(E, Wih, Whh, bih, bhh, word, hprev, hnew, merge);
    k_q  <<<128, 256, 0, stream>>>(Wq, hnew, qbuf);
    k_attn<<<1, 256, 0, stream>>>(emb, qbuf, merge, wts + (size_t)t * Ss);

    float* logits = lp + (size_t)t * Vv;          // raw logits written in place
    int groups = (Vv + 15) / 16;                  // 3142 row-groups
    int blocks = (groups + 7) / 8;                // 8 waves (groups) per block
    k_wout_wmma<<<blocks, 256, 0, stream>>>(Wout, bout, merge, logits);

    k_finalize<<<1, 1024, 0, stream>>>(logits, toks + t, word);
  }
}